// AGCNN_88802743812388
// MI455X (gfx1250) — compile-verified
//
#include <hip/hip_runtime.h>
#include <math.h>

// ---------------------------------------------------------------------------
// AGCNN (DGCNN-style) forward for MI455X / gfx1250.
// fp32 WMMA (V_WMMA_F32_16X16X4_F32) for Gram matrices and 1x1 convs.
// wave32; 256-thread blocks = 8 waves. All K dims padded to multiples of 4
// so the WMMA inner loops are branch-free (no divergent exec-mask loads).
// ---------------------------------------------------------------------------

#define NPTS   2048
#define NBATCH 16
#define KNN    20
#define BCHUNK 4            // batches of distance matrix staged at once (64 MB)

typedef __attribute__((ext_vector_type(2))) float v2f;
typedef __attribute__((ext_vector_type(8))) float v8f;

__device__ __forceinline__ v8f wmma_f32_k4(v2f a, v2f b, v8f c) {
  // D(16x16,f32) = A(16x4,f32) * B(4x16,f32) + C
  return __builtin_amdgcn_wmma_f32_16x16x4_f32(
      /*neg_a=*/false, a, /*neg_b=*/false, b,
      /*c_mod=*/(short)0, c, /*reuse_a=*/false, /*reuse_b=*/false);
}

// ------------------------------ padding ------------------------------------
// (B,3,N) -> (B,4,N), channel 3 zeroed (so dist GEMM runs with K=4).
__global__ __launch_bounds__(256) void agcnn_pad_x(
    const float* __restrict__ x, float* __restrict__ xp) {
  int i = blockIdx.x * 256 + threadIdx.x;          // B*4*N total
  int n = i & (NPTS - 1);
  int r = i >> 11;
  int c = r & 3;
  int b = r >> 2;
  xp[i] = (c < 3) ? x[((size_t)b * 3 + c) * NPTS + n] : 0.f;
}

// W1 (64,6) -> (64,8) zero padded.
__global__ void agcnn_pad_w1(const float* __restrict__ W,
                             float* __restrict__ Wp) {
  int i = blockIdx.x * 256 + threadIdx.x;          // 512 total
  if (i >= 64 * 8) return;
  int c = i & 7, o = i >> 3;
  Wp[i] = (c < 6) ? W[o * 6 + c] : 0.f;
}

// --------------------------- squared norms ---------------------------------
__global__ __launch_bounds__(256) void agcnn_sqnorm(
    const float* __restrict__ x, float* __restrict__ xx, int D, long bstr) {
  int i = blockIdx.x * 256 + threadIdx.x;          // i in [0, B*N)
  int n = i & (NPTS - 1);
  int b = i >> 11;
  const float* xb = x + (size_t)b * bstr;
  float s = 0.f;
  for (int d = 0; d < D; ++d) {
    float v = xb[(size_t)d * NPTS + n];
    s += v * v;
  }
  xx[i] = s;
}

// --------------------------- distance GEMM ---------------------------------
// dist[cb][n][m] = 2 * <x_n, x_m> - ||x_n||^2 - ||x_m||^2
// Wave tile: 16(rows) x 64(cols). Row-tile index is fastest within a block so
// all 8 waves of a block share identical B fragments (L0 hits). D % 4 == 0.
__global__ __launch_bounds__(256) void agcnn_dist(
    const float* __restrict__ x, const float* __restrict__ xx,
    float* __restrict__ dist, int D, long bstr, int b_off, int bc) {
  const int TR = NPTS / 16, TC = NPTS / 64;
  int wave = threadIdx.x >> 5, lane = threadIdx.x & 31;
  int lm = lane & 15, hs = lane >> 4;
  int t = blockIdx.x * 8 + wave;
  if (t >= bc * TR * TC) return;
  int cb = t / (TR * TC);
  int rem = t % (TR * TC);
  int ct = rem / TR;                    // shared by 8 consecutive waves
  int rt = rem % TR;
  int gb = b_off + cb;
  int n0 = rt * 16, m0 = ct * 64;
  const float* xb = x + (size_t)gb * bstr;
  const float* pn = xb + n0 + lm;       // A fragment base
  const float* pm = xb + m0 + lm;       // B fragment base

  v8f c0 = {}, c1 = {}, c2 = {}, c3 = {};
  for (int d0 = 0; d0 < D; d0 += 4) {
    size_t ro = (size_t)(d0 + 2 * hs) * NPTS;
    v2f a;  a[0]  = pn[ro];      a[1]  = pn[ro + NPTS];
    v2f b0; b0[0] = pm[ro];      b0[1] = pm[ro + NPTS];
    v2f b1; b1[0] = pm[ro + 16]; b1[1] = pm[ro + NPTS + 16];
    v2f b2; b2[0] = pm[ro + 32]; b2[1] = pm[ro + NPTS + 32];
    v2f b3; b3[0] = pm[ro + 48]; b3[1] = pm[ro + NPTS + 48];
    c0 = wmma_f32_k4(a, b0, c0);
    c1 = wmma_f32_k4(a, b1, c1);
    c2 = wmma_f32_k4(a, b2, c2);
    c3 = wmma_f32_k4(a, b3, c3);
  }

  const float* xxb = xx + (size_t)gb * NPTS;
  float xr[8];
#pragma unroll
  for (int r = 0; r < 8; ++r) xr[r] = xxb[n0 + r + 8 * hs];
  float* dbase = dist + (size_t)cb * NPTS * NPTS;

#define DIST_EPI(ACC, CC)                                              \
  do {                                                                 \
    int col = m0 + 16 * (CC) + lm;                                     \
    float xc = xxb[col];                                               \
    float* dc = dbase + col;                                           \
    _Pragma("unroll") for (int r = 0; r < 8; ++r) {                    \
      dc[(size_t)(n0 + r + 8 * hs) * NPTS] = 2.f * ACC[r] - xr[r] - xc;\
    }                                                                  \
  } while (0)

  DIST_EPI(c0, 0);
  DIST_EPI(c1, 1);
  DIST_EPI(c2, 2);
  DIST_EPI(c3, 3);
#undef DIST_EPI
}

// ------------------ top-k + softmax + gather + aggregate -------------------
// One wave per (cb, n). Lane-local sorted top-20 over stride-32 scan, then
// 20-round shuffle argmax merge (ties -> smaller index, matching lax.top_k).
__global__ __launch_bounds__(256) void agcnn_topk_agg(
    const float* __restrict__ x, const float* __restrict__ dist,
    float* __restrict__ out, int D, long in_bstr, long out_bstr,
    int b_off, int bc) {
  int wave = threadIdx.x >> 5, lane = threadIdx.x & 31;
  int t = blockIdx.x * 8 + wave;
  if (t >= bc * NPTS) return;
  int cb = t / NPTS, n = t % NPTS;
  int gb = b_off + cb;
  const float* drow = dist + ((size_t)cb * NPTS + n) * NPTS;

  float val[KNN];
  int   idx[KNN];
#pragma unroll
  for (int i = 0; i < KNN; ++i) { val[i] = -3.4e38f; idx[i] = 0x7fffffff; }

  for (int j = lane; j < NPTS; j += 32) {
    __builtin_prefetch(drow + j + 256, 0, 1);      // global_prefetch_b8
    float v = drow[j];
    if (v > val[KNN - 1]) {
      int p = KNN - 1;
      while (p > 0 && v > val[p - 1]) {
        val[p] = val[p - 1]; idx[p] = idx[p - 1]; --p;
      }
      val[p] = v; idx[p] = j;
    }
  }

  float sv[KNN];
  int   si[KNN];
  int p = 0;
#pragma unroll
  for (int tt = 0; tt < KNN; ++tt) {
    float cv = (p < KNN) ? val[p] : -3.4e38f;
    int   ci = (p < KNN) ? idx[p] : 0x7fffffff;
    for (int off = 16; off >= 1; off >>= 1) {
      float ov = __shfl_xor(cv, off, 32);
      int   oi = __shfl_xor(ci, off, 32);
      if (ov > cv || (ov == cv && oi < ci)) { cv = ov; ci = oi; }
    }
    sv[tt] = cv; si[tt] = ci;
    if (p < KNN && val[p] == cv && idx[p] == ci) ++p;   // winning lane pops
  }

  float m = sv[0];
  float w[KNN];
  float s = 0.f;
#pragma unroll
  for (int tt = 0; tt < KNN; ++tt) { w[tt] = expf(sv[tt] - m); s += w[tt]; }
  float inv = 1.f / s;

  const float* xb = x + (size_t)gb * in_bstr;
  float* ob = out + (size_t)gb * out_bstr;
  for (int d = lane; d < D; d += 32) {
    const float* xd = xb + (size_t)d * NPTS;
    float center = xd[n];
    float acc = 0.f;
#pragma unroll
    for (int tt = 0; tt < KNN; ++tt) acc += w[tt] * (xd[si[tt]] - center);
    ob[(size_t)d * NPTS + n]       = center;       // passthrough channels
    ob[(size_t)(d + D) * NPTS + n] = acc * inv;    // aggregated channels
  }
}

// ------------------------------ 1x1 conv GEMM ------------------------------
// y[b][o][n] = sum_c W[o][c] * in[b][c][n]; wave tile 16(o) x 64(n).
// o-tile index fastest within a block (waves share activation fragments).
// C % 4 == 0. Deterministic BN partials (half-wave shuffle reduce).
__global__ __launch_bounds__(256) void agcnn_conv(
    const float* __restrict__ in, const float* __restrict__ W,
    float* __restrict__ y, float* __restrict__ ps, float* __restrict__ psq,
    int C, int O, long bstr) {
  const int TC = NPTS / 64;
  int wave = threadIdx.x >> 5, lane = threadIdx.x & 31;
  int lm = lane & 15, hs = lane >> 4;
  int t = blockIdx.x * 8 + wave;
  int OT = O / 16;
  int per_b = OT * TC;
  if (t >= NBATCH * per_b) return;
  int b = t / per_b;
  int rem = t % per_b;
  int ct = rem / OT;
  int ot = rem % OT;                    // fastest -> shared B frags in block
  int o0 = ot * 16, n0 = ct * 64;
  const float* inb = in + (size_t)b * bstr + n0 + lm;
  const float* wr = W + (size_t)(o0 + lm) * C + 2 * hs;

  v8f c0 = {}, c1 = {}, c2 = {}, c3 = {};
  for (int k0 = 0; k0 < C; k0 += 4) {
    size_t ro = (size_t)(k0 + 2 * hs) * NPTS;
    v2f a;  a[0]  = wr[k0];        a[1]  = wr[k0 + 1];
    v2f b0; b0[0] = inb[ro];       b0[1] = inb[ro + NPTS];
    v2f b1; b1[0] = inb[ro + 16];  b1[1] = inb[ro + NPTS + 16];
    v2f b2; b2[0] = inb[ro + 32];  b2[1] = inb[ro + NPTS + 32];
    v2f b3; b3[0] = inb[ro + 48];  b3[1] = inb[ro + NPTS + 48];
    c0 = wmma_f32_k4(a, b0, c0);
    c1 = wmma_f32_k4(a, b1, c1);
    c2 = wmma_f32_k4(a, b2, c2);
    c3 = wmma_f32_k4(a, b3, c3);
  }

  float* yb = y + (size_t)b * O * NPTS;
  int P = NBATCH * TC;                             // partials per channel
#pragma unroll
  for (int r = 0; r < 8; ++r) {
    int o = o0 + r + 8 * hs;
    float* yo = yb + (size_t)o * NPTS + n0 + lm;
    float v0 = c0[r], v1 = c1[r], v2 = c2[r], v3 = c3[r];
    yo[0]  = v0;
    yo[16] = v1;
    yo[32] = v2;
    yo[48] = v3;
    float s = v0 + v1 + v2 + v3;
    float q = v0 * v0 + v1 * v1 + v2 * v2 + v3 * v3;
    for (int off = 8; off >= 1; off >>= 1) {       // reduce 16-lane half
      s += __shfl_xor(s, off, 32);
      q += __shfl_xor(q, off, 32);
    }
    if (lm == 0) {
      size_t pi = (size_t)o * P + (size_t)b * TC + ct;
      ps[pi]  = s;
      psq[pi] = q;
    }
  }
}

// Fixed-order reduction of BN partials -> per-channel sum / sumsq.
__global__ void agcnn_bn_reduce(const float* __restrict__ ps,
                                const float* __restrict__ psq,
                                float* __restrict__ sum,
                                float* __restrict__ sumsq, int O, int P) {
  int o = blockIdx.x * blockDim.x + threadIdx.x;
  if (o >= O) return;
  float s = 0.f, q = 0.f;
  for (int i = 0; i < P; ++i) {
    s += ps[(size_t)o * P + i];
    q += psq[(size_t)o * P + i];
  }
  sum[o] = s;
  sumsq[o] = q;
}

// BN (training stats) + LeakyReLU, written to dst (and optional dst2).
__global__ __launch_bounds__(256) void agcnn_bn_lrelu(
    const float* __restrict__ y, const float* __restrict__ sum,
    const float* __restrict__ sumsq, const float* __restrict__ gamma,
    const float* __restrict__ beta, float* __restrict__ dst, long dstr,
    float* __restrict__ dst2, long dstr2, int O) {
  long i = (long)blockIdx.x * 256 + threadIdx.x;   // i in [0, B*O*N)
  int n = (int)(i & (NPTS - 1));
  long r = i >> 11;
  int o = (int)(r % O);
  int b = (int)(r / O);
  const float invBN = 1.f / (float)(NBATCH * NPTS);
  float mean = sum[o] * invBN;
  float var = sumsq[o] * invBN - mean * mean;
  float sc = gamma[o] * rsqrtf(var + 1e-5f);
  float sh = beta[o] - mean * sc;
  float v = y[i] * sc + sh;
  v = (v >= 0.f) ? v : 0.01f * v;
  size_t di = (size_t)b * dstr + (size_t)o * NPTS + n;
  dst[di] = v;
  if (dst2) dst2[(size_t)b * dstr2 + (size_t)o * NPTS + n] = v;
}

// ------------------------------- host side ---------------------------------
extern "C" void kernel_launch(void* const* d_in, const int* in_sizes, int n_in,
                              void* d_out, int out_size, void* d_ws,
                              size_t ws_size, hipStream_t stream) {
  (void)in_sizes; (void)n_in; (void)out_size; (void)ws_size;
  const float* x = (const float*)d_in[0];
  const float* Wl[5] = {(const float*)d_in[1], (const float*)d_in[2],
                        (const float*)d_in[3], (const float*)d_in[4],
                        (const float*)d_in[5]};
  const float* gm[5] = {(const float*)d_in[6],  (const float*)d_in[8],
                        (const float*)d_in[10], (const float*)d_in[12],
                        (const float*)d_in[14]};
  const float* bt[5] = {(const float*)d_in[7],  (const float*)d_in[9],
                        (const float*)d_in[11], (const float*)d_in[13],
                        (const float*)d_in[15]};

  float* ws = (float*)d_ws;
  // dist (BCHUNK*N*N) and y (B*512*N) are both exactly 16,777,216 floats and
  // have disjoint lifetimes -> alias at offset 0.
  float* dist  = ws;
  float* yb    = ws;
  float* cat   = ws + (size_t)16777216;                 // (B,512,N) concat
  float* gsain = cat + (size_t)NBATCH * 512 * NPTS;     // (B,<=256,N) gsa out
  float* xxb   = gsain + (size_t)NBATCH * 256 * NPTS;   // (B,N) sq norms
  float* ps    = xxb + (size_t)NBATCH * NPTS;           // (512, B*32) partials
  float* psq   = ps + (size_t)512 * NBATCH * (NPTS / 64);
  float* sum   = psq + (size_t)512 * NBATCH * (NPTS / 64);
  float* sumsq = sum + 512;
  float* xp    = sumsq + 512;                           // (B,4,N) padded x
  float* Wp    = xp + (size_t)NBATCH * 4 * NPTS;        // (64,8) padded W1

  float* out5 = (float*)d_out;                          // (B,512,N)
  float* out3 = out5 + (size_t)NBATCH * 512 * NPTS;     // (B,128,N)

  auto run_gsa = [&](const float* in, int D, int Dpad, long bstr, float* outp,
                     long out_bstr) {
    agcnn_sqnorm<<<(NBATCH * NPTS) / 256, 256, 0, stream>>>(in, xxb, Dpad,
                                                           bstr);
    for (int b0 = 0; b0 < NBATCH; b0 += BCHUNK) {
      int waves_d = BCHUNK * (NPTS / 16) * (NPTS / 64);
      agcnn_dist<<<waves_d / 8, 256, 0, stream>>>(in, xxb, dist, Dpad, bstr,
                                                  b0, BCHUNK);
      int waves_t = BCHUNK * NPTS;
      agcnn_topk_agg<<<waves_t / 8, 256, 0, stream>>>(in, dist, outp, D, bstr,
                                                      out_bstr, b0, BCHUNK);
    }
  };

  auto run_conv = [&](const float* in, long bstr, int C, int O, const float* W,
                      const float* g, const float* b, float* dst, long dstr,
                      float* dst2, long dstr2) {
    int waves = NBATCH * (O / 16) * (NPTS / 64);
    agcnn_conv<<<waves / 8, 256, 0, stream>>>(in, W, yb, ps, psq, C, O, bstr);
    agcnn_bn_reduce<<<(O + 63) / 64, 64, 0, stream>>>(
        ps, psq, sum, sumsq, O, NBATCH * (NPTS / 64));
    long total = (long)NBATCH * O * NPTS;
    agcnn_bn_lrelu<<<(int)(total / 256), 256, 0, stream>>>(
        yb, sum, sumsq, g, b, dst, dstr, dst2, dstr2, O);
  };

  // Padding prep: x -> (B,4,N); W1 -> (64,8); gsain ch 6..7 zeroed for conv1.
  agcnn_pad_x<<<(NBATCH * 4 * NPTS) / 256, 256, 0, stream>>>(x, xp);
  agcnn_pad_w1<<<2, 256, 0, stream>>>(Wl[0], Wp);
  hipMemsetAsync(gsain, 0, (size_t)NBATCH * 8 * NPTS * sizeof(float), stream);

  // Layer 1: gsa(x, D=3, K padded to 4) -> (B,8,N); conv 8->64 -> cat[:,0:64]
  run_gsa(xp, 3, 4, 4L * NPTS, gsain, 8L * NPTS);
  run_conv(gsain, 8L * NPTS, 8, 64, Wp, gm[0], bt[0],
           cat, 512L * NPTS, nullptr, 0);
  // Layer 2: gsa(x1, D=64) -> (B,128,N); conv 128->64 -> x2 = cat[:, 64:128]
  run_gsa(cat, 64, 64, 512L * NPTS, gsain, 128L * NPTS);
  run_conv(gsain, 128L * NPTS, 128, 64, Wl[1], gm[1], bt[1],
           cat + 64L * NPTS, 512L * NPTS, nullptr, 0);
  // Layer 3: gsa(x2) -> (B,128,N); conv 128->128 -> x3 = cat[:,128:256] + out3
  run_gsa(cat + 64L * NPTS, 64, 64, 512L * NPTS, gsain, 128L * NPTS);
  run_conv(gsain, 128L * NPTS, 128, 128, Wl[2], gm[2], bt[2],
           cat + 128L * NPTS, 512L * NPTS, out3, 128L * NPTS);
  // Layer 4: gsa(x3, D=128) -> (B,256,N); conv 256->256 -> x4 = cat[:,256:512]
  run_gsa(cat + 128L * NPTS, 128, 128, 512L * NPTS, gsain, 256L * NPTS);
  run_conv(gsain, 256L * NPTS, 256, 256, Wl[3], gm[3], bt[3],
           cat + 256L * NPTS, 512L * NPTS, nullptr, 0);
  // Layer 5: conv 512->512 over cat -> x5 (d_out)
  run_conv(cat, 512L * NPTS, 512, 512, Wl[4], gm[4], bt[4],
           out5, 512L * NPTS, nullptr, 0);
}